// Sampler_8186207666693
// MI455X (gfx1250) — compile-verified
//
#include <hip/hip_runtime.h>
#include <stdint.h>

// ---------------- tunables / problem constants ----------------
#define TPB      512           // 16 wave32 waves per workgroup
#define NWAVES   (TPB / 32)
#define MAXV     128000
#define MASKW    (MAXV / 32)   // 4000 words = 16KB history bitmask
#define NBINS    4096          // radix histogram: top 12 bits of sortable key
#define NCHUNK   (NBINS / 8)   // 512 chunk sums
#define CANDMAX  512
#define TOPBUF   64

#define RP       1.35f
#define INV_RP   (1.0f / 1.35f)
#define TOPP     0.8f
#define INV_T    1.25f         // 1 / TEMPERATURE(0.8)

typedef float v4f __attribute__((ext_vector_type(4)));
typedef int   v4i __attribute__((ext_vector_type(4)));

#if defined(__HIP_DEVICE_COMPILE__) && \
    __has_builtin(__builtin_amdgcn_global_load_async_to_lds_b128)
#define USE_ASYNC_LDS 1
// exact param types per clang diagnostic: (v4i __device__*, v4i __shared__*, imm, imm)
typedef __attribute__((address_space(1))) v4i* gptr128_t;
typedef __attribute__((address_space(3))) v4i* lptr128_t;
#endif

__device__ __forceinline__ void wait_async_le1() {
#if __has_builtin(__builtin_amdgcn_s_wait_asynccnt)
  __builtin_amdgcn_s_wait_asynccnt(1);
#else
  asm volatile("s_wait_asynccnt 1" ::: "memory");
#endif
}
__device__ __forceinline__ void wait_async_le0() {
#if __has_builtin(__builtin_amdgcn_s_wait_asynccnt)
  __builtin_amdgcn_s_wait_asynccnt(0);
#else
  asm volatile("s_wait_asynccnt 0" ::: "memory");
#endif
}
__device__ __forceinline__ void wait_store0() {
#if __has_builtin(__builtin_amdgcn_s_wait_storecnt)
  __builtin_amdgcn_s_wait_storecnt(0);
#else
  asm volatile("s_wait_storecnt 0" ::: "memory");
#endif
}

// monotonic float -> uint key (total order)
__device__ __forceinline__ uint32_t fkey(float x) {
  uint32_t b = __float_as_uint(x);
  uint32_t m = (uint32_t)((int32_t)b >> 31) | 0x80000000u;
  return b ^ m;
}

__device__ __forceinline__ float penal(float v, const uint32_t* mask, int idx) {
  uint32_t bit = (mask[idx >> 5] >> (idx & 31)) & 1u;
  if (bit) v = (v < 0.0f) ? v * RP : v * INV_RP;
  return v;
}

__global__ __launch_bounds__(TPB) void sampler_kernel(
    const float* __restrict__ logits,
    const long long* __restrict__ prev,
    const float* __restrict__ uin,
    const int* __restrict__ topk_p,
    int* __restrict__ samples,
    float* __restrict__ probs,
    int V, int H)
{
  __shared__ uint32_t mask[MASKW];
  __shared__ uint32_t hist[NBINS];
  __shared__ uint32_t csum[NCHUNK];
  __shared__ float    cand_val[CANDMAX];
  __shared__ int      cand_idx[CANDMAX];
  __shared__ float    sval[TOPBUF];
  __shared__ int      sidx[TOPBUF];
  __shared__ float    wredf[NWAVES];
  __shared__ uint32_t s_keylo;
  __shared__ unsigned s_cand_cnt;
  __shared__ float    s_M, s_Z, s_sum2, s_top;
  __shared__ int      s_K, s_kk;
#ifdef USE_ASYNC_LDS
  __shared__ v4f      stage[NWAVES][2][32];
#endif

  const int tid  = threadIdx.x;
  const int lane = tid & 31;
  const int wv   = tid >> 5;
  const int r    = blockIdx.x;

  const float* row  = logits + (size_t)r * V;
  const float* urow = uin    + (size_t)r * V;
  float*       prow = probs  + (size_t)r * V;

  // ---- 0) zero LDS, read k ----
  for (int i = tid; i < MASKW; i += TPB) mask[i] = 0u;
  for (int i = tid; i < NBINS; i += TPB) hist[i] = 0u;
  if (tid == 0) {
    s_cand_cnt = 0u;
    int kk = topk_p[0];
    if (kk < 1) kk = 1;
    if (kk > TOPBUF) kk = TOPBUF;
    s_kk = kk;
  }
  __syncthreads();

  // ---- 1) history bitmask (repetition penalty positions) ----
  for (int i = tid; i < H; i += TPB) {
    int ti = (int)prev[(size_t)r * H + i];
    if (ti >= 0 && ti < V) atomicOr(&mask[ti >> 5], 1u << (ti & 31));
  }
  __syncthreads();

  // ---- 2) pass 1 (HBM stream): max + histogram + NT zero-fill probs ----
  float tmax = -3.4e38f;
  for (int base = tid * 4; base + 3 < V; base += TPB * 4) {
    __builtin_prefetch(row + base + TPB * 8, 0, 0);          // global_prefetch_b8
    v4f x = *(const v4f*)(row + base);
    v4f z = {0.f, 0.f, 0.f, 0.f};
    __builtin_nontemporal_store(z, (v4f*)(prow + base));     // NT store_b128
#pragma unroll
    for (int j = 0; j < 4; ++j) {
      float v = penal(x[j], mask, base + j);
      tmax = fmaxf(tmax, v);
      atomicAdd(&hist[fkey(v) >> 20], 1u);
    }
  }
  for (int i = (V & ~3) + tid; i < V; i += TPB) {            // generic tail
    float v = penal(row[i], mask, i);
    prow[i] = 0.f;
    tmax = fmaxf(tmax, v);
    atomicAdd(&hist[fkey(v) >> 20], 1u);
  }
  wait_store0();  // all zero stores at coherency point before later scatter

  // block max reduction
  for (int off = 16; off > 0; off >>= 1)
    tmax = fmaxf(tmax, __shfl_xor(tmax, off, 32));
  if (lane == 0) wredf[wv] = tmax;
  __syncthreads();
  if (tid == 0) {
    float m = -3.4e38f;
    for (int i = 0; i < NWAVES; ++i) m = fmaxf(m, wredf[i]);
    s_M = m;
  }
  __syncthreads();
  const float M = s_M;

  // ---- 3) histogram scan: find bin containing the k-th largest ----
  for (int c = tid; c < NCHUNK; c += TPB) {
    uint32_t s = 0;
#pragma unroll
    for (int k = 0; k < 8; ++k) s += hist[c * 8 + k];
    csum[c] = s;
  }
  __syncthreads();
  if (wv == 0) {
    const uint32_t NEED = (uint32_t)s_kk;
    uint32_t sup = 0;
    for (int k = 0; k < 16; ++k) sup += csum[lane * 16 + k];
    uint32_t acc = sup;                      // inclusive suffix sum over lanes
    for (int off = 1; off < 32; off <<= 1) {
      uint32_t o = __shfl_down(acc, off, 32);
      if (lane + off < 32) acc += o;
    }
    uint32_t above = acc - sup;              // count in higher lanes (higher bins)
    if (above < NEED && above + sup >= NEED) {
      uint32_t rem = NEED - above;
      for (int c = lane * 16 + 15; c >= lane * 16; --c) {
        uint32_t cs = csum[c];
        if (cs >= rem) {
          for (int b = c * 8 + 7; b >= c * 8; --b) {
            uint32_t hb = hist[b];
            if (hb >= rem) { s_keylo = ((uint32_t)b) << 20; break; }
            rem -= hb;
          }
          break;
        }
        rem -= cs;
      }
    }
  }
  __syncthreads();
  const uint32_t keylo = s_keylo;

  // ---- 4) pass 2 (L2-resident): Z = sum exp(v-M), collect candidates ----
  float zsum = 0.f;
  const int nch = V >> 7;  // 128-element chunks (one per wave iteration)

#ifdef USE_ASYNC_LDS
  {
    int buf = 0;
    if (wv < nch) {
      __builtin_amdgcn_global_load_async_to_lds_b128(
          (gptr128_t)(row + (size_t)wv * 128 + lane * 4),
          (lptr128_t)&stage[wv][0][lane], 0, 0);
    }
    for (int c = wv; c < nch; c += NWAVES) {
      int cn = c + NWAVES;
      if (cn < nch) {
        __builtin_amdgcn_global_load_async_to_lds_b128(
            (gptr128_t)(row + (size_t)cn * 128 + lane * 4),
            (lptr128_t)&stage[wv][buf ^ 1][lane], 0, 0);
        wait_async_le1();   // previous buffer complete (in-order completion)
      } else {
        wait_async_le0();
      }
      __asm__ __volatile__("" ::: "memory");
      v4f x = stage[wv][buf][lane];           // ds_load_b128
      int gbase = c * 128 + lane * 4;
#pragma unroll
      for (int j = 0; j < 4; ++j) {
        float v = penal(x[j], mask, gbase + j);
        zsum += __expf(v - M);
        if (fkey(v) >= keylo) {
          unsigned pos = atomicAdd(&s_cand_cnt, 1u);
          if (pos < CANDMAX) { cand_val[pos] = v; cand_idx[pos] = gbase + j; }
        }
      }
      buf ^= 1;
    }
  }
#else
  for (int c = wv; c < nch; c += NWAVES) {
    int gbase = c * 128 + lane * 4;
    v4f x = *(const v4f*)(row + gbase);
#pragma unroll
    for (int j = 0; j < 4; ++j) {
      float v = penal(x[j], mask, gbase + j);
      zsum += __expf(v - M);
      if (fkey(v) >= keylo) {
        unsigned pos = atomicAdd(&s_cand_cnt, 1u);
        if (pos < CANDMAX) { cand_val[pos] = v; cand_idx[pos] = gbase + j; }
      }
    }
  }
#endif
  for (int i = (nch << 7) + tid; i < V; i += TPB) {  // generic tail
    float v = penal(row[i], mask, i);
    zsum += __expf(v - M);
    if (fkey(v) >= keylo) {
      unsigned pos = atomicAdd(&s_cand_cnt, 1u);
      if (pos < CANDMAX) { cand_val[pos] = v; cand_idx[pos] = i; }
    }
  }

  // block sum reduction for Z
  for (int off = 16; off > 0; off >>= 1)
    zsum += __shfl_xor(zsum, off, 32);
  if (lane == 0) wredf[wv] = zsum;
  __syncthreads();
  if (tid == 0) {
    float s = 0.f;
    for (int i = 0; i < NWAVES; ++i) s += wredf[i];
    s_Z = s;
  }
  __syncthreads();

  // ---- 5) stable rank-select of the top TOPBUF candidates ----
  const int n = (s_cand_cnt < (unsigned)CANDMAX) ? (int)s_cand_cnt : CANDMAX;
  for (int i = tid; i < n; i += TPB) {
    float vi = cand_val[i];
    int   ii = cand_idx[i];
    int   rk = 0;
    for (int j = 0; j < n; ++j) {
      float vj = cand_val[j];
      rk += (vj > vi) || (vj == vi && cand_idx[j] < ii);  // argsort stability
    }
    if (rk < TOPBUF) { sval[rk] = vi; sidx[rk] = ii; }
  }
  __syncthreads();

  // ---- 6) nucleus (top-p) + top-k pivot + temperature softmax norm ----
  if (tid == 0) {
    int kk = s_kk;
    if (kk > n) kk = n;
    float invZ = 1.0f / s_Z;
    float c = 0.f;
    int m = kk;
    for (int i = 0; i < kk; ++i) {
      c += __expf(sval[i] - M) * invZ;     // cumulative softmax (full-V denom)
      if (i > 0 && c > TOPP) { m = i; break; }
    }
    int K = m;                             // kept support size (<= kk <= 64)
    float top = sval[0];
    float s2 = 0.f;
    for (int i = 0; i < K; ++i) s2 += __expf((sval[i] - top) * INV_T);
    s_K = K; s_sum2 = s2; s_top = top;
  }
  __syncthreads();

  // ---- 7) scatter probs, gather u only at kept tokens, wave argmax ----
  if (wv == 0) {
    const int   K    = s_K;
    const float top  = s_top;
    const float inv2 = 1.0f / s_sum2;
    float bestr = -1.f;
    int   besti = 0x7fffffff;
    for (int i = lane; i < K; i += 32) {
      float p  = __expf((sval[i] - top) * INV_T) * inv2;
      int   idx = sidx[i];
      prow[idx] = p;                        // overwrite the zero written in pass 1
      float uu = urow[idx];
      float q  = fmaxf(-log1pf(-uu), 1e-10f);
      float ratio = p / q;
      if (ratio > bestr || (ratio == bestr && idx < besti)) { bestr = ratio; besti = idx; }
    }
    for (int off = 16; off > 0; off >>= 1) {
      float rr = __shfl_xor(bestr, off, 32);
      int   ri = __shfl_xor(besti, off, 32);
      if (rr > bestr || (rr == bestr && ri < besti)) { bestr = rr; besti = ri; }
    }
    if (lane == 0) samples[r] = (besti == 0x7fffffff) ? 0 : besti;
  }
}

extern "C" void kernel_launch(void* const* d_in, const int* in_sizes, int n_in,
                              void* d_out, int out_size, void* d_ws, size_t ws_size,
                              hipStream_t stream) {
  const float*     logits = (const float*)d_in[0];
  const long long* prev   = (const long long*)d_in[1];   // int64 history tokens
  const float*     uu     = (const float*)d_in[2];
  const int*       topk   = (const int*)d_in[3];

  const int V = 128000;
  const int B = in_sizes[0] / V;
  const int H = (B > 0) ? in_sizes[1] / B : 0;

  // d_out layout: samples (B int32) followed by probs (B*V float32)
  int*   samples = (int*)d_out;
  float* probs   = (float*)d_out + B;

  sampler_kernel<<<B, TPB, 0, stream>>>(logits, prev, uu, topk,
                                        samples, probs, V, H);
}